// LMHead_48627619725771
// MI455X (gfx1250) — compile-verified
//
#include <hip/hip_runtime.h>

#define D_MODEL 1024
#define SEQ     2048
#define BATCH   4
#define VOCAB   50257
#define NTILES  ((VOCAB + 15) / 16)   // 3142
#define LDS_STRIDE 1032               // 4128 B row stride: breaks bank conflicts vs 4096

typedef __attribute__((ext_vector_type(2))) float v2f;
typedef __attribute__((ext_vector_type(8))) float v8f;

// ---------------------------------------------------------------------------
// Stage 1: partial seq-sum of x. grid = (4 dchunks, 32 schunks, 4 batches).
// part[sc*4096 + b*1024 + d] = sum over 64 consecutive s of x[b][s][d].
// Coalesced along d; 512 blocks saturate HBM for the 33.5 MB of x.
// ---------------------------------------------------------------------------
__global__ __launch_bounds__(256) void lmhead_reduce_partial(
    const float* __restrict__ x, float* __restrict__ part) {
  const int d  = blockIdx.x * 256 + threadIdx.x;   // 0..1023
  const int sc = blockIdx.y;                        // 0..31
  const int b  = blockIdx.z;                        // 0..3
  const float* p = x + ((size_t)b * SEQ + (size_t)sc * 64) * D_MODEL + d;
  float s = 0.0f;
#pragma unroll 8
  for (int i = 0; i < 64; ++i) s += p[(size_t)i * D_MODEL];
  part[((size_t)sc * (BATCH * D_MODEL)) + b * D_MODEL + d] = s;
}

// ---------------------------------------------------------------------------
// Stage 2: fold 32 partials -> xsum[4*1024] (16 KB, lives in L2 afterwards).
// ---------------------------------------------------------------------------
__global__ __launch_bounds__(256) void lmhead_combine(
    const float* __restrict__ part, float* __restrict__ xsum) {
  const int i = blockIdx.x * 256 + threadIdx.x;    // 0..4095
  float s = 0.0f;
#pragma unroll
  for (int j = 0; j < 32; ++j) s += part[(size_t)j * (BATCH * D_MODEL) + i];
  xsum[i] = s;
}

// ---------------------------------------------------------------------------
// Stage 3: out[b, v] = xsum[b,:] . W[v,:] + SEQ * bias[v] via f32 WMMA.
// One wave per 16-wide vocab tile; A = xsum padded to 16 rows (rows 4..15
// zeroed by value predicate, EXEC stays all-ones as WMMA requires).
// Memory-bound on W: each v_wmma_f32_16x16x4_f32 consumes exactly the 256 B
// of W its wave just loaded (float2 per lane, fully coalesced per row stream).
// ---------------------------------------------------------------------------
__global__ __launch_bounds__(256) void lmhead_gemv_wmma(
    const float* __restrict__ W, const float* __restrict__ bias,
    const float* __restrict__ xsum, float* __restrict__ out) {
  __shared__ float lx[BATCH * LDS_STRIDE];

  const int tid = threadIdx.x;
  // Cache xsum (16 KB) in LDS with padded stride.
  for (int i = tid; i < BATCH * D_MODEL; i += 256)
    lx[(i >> 10) * LDS_STRIDE + (i & (D_MODEL - 1))] = xsum[i];
  __syncthreads();

  const int wave = tid >> 5;
  const int lane = tid & 31;
  const int tile = blockIdx.x * 8 + wave;
  if (tile >= NTILES) return;                       // wave-uniform exit

  const int v0  = tile * 16;
  const int n   = lane & 15;                        // N (and A-row M) for this lane
  const int klo = (lane >> 4) * 2;                  // K pair: 0,1 (low half) / 2,3 (high half)
  const int col = v0 + n;
  const int row = col < VOCAB ? col : (VOCAB - 1);  // clamp OOB rows of last tile
  const float* wr = W + (size_t)row * D_MODEL;

  const bool  vm = (n < BATCH);                     // only rows 0..3 of A are real
  const float* ar = lx + (n & (BATCH - 1)) * LDS_STRIDE;

  v8f c = {};
#pragma unroll 8
  for (int k = 0; k < D_MODEL; k += 4) {
    v2f bfrag = *(const v2f*)(wr + k + klo);        // B[k..k+3][n] fragment
    v2f t     = *(const v2f*)(ar + k + klo);        // A[m][k..k+3] fragment
    v2f afrag;
    afrag.x = vm ? t.x : 0.0f;
    afrag.y = vm ? t.y : 0.0f;
    // D = A(16x4) x B(4x16) + C ; full f32 precision
    c = __builtin_amdgcn_wmma_f32_16x16x4_f32(
        /*neg_a=*/false, afrag, /*neg_b=*/false, bfrag,
        /*c_mod=*/(short)0, c, /*reuse_a=*/false, /*reuse_b=*/false);
  }

  // C/D layout: lanes 0..15 hold N=lane, M = VGPR index (0..7). Batch rows 0..3.
  if (lane < 16 && col < VOCAB) {
    const float bb = (float)SEQ * bias[col];
    out[(size_t)0 * VOCAB + col] = c[0] + bb;
    out[(size_t)1 * VOCAB + col] = c[1] + bb;
    out[(size_t)2 * VOCAB + col] = c[2] + bb;
    out[(size_t)3 * VOCAB + col] = c[3] + bb;
  }
}

// ---------------------------------------------------------------------------
extern "C" void kernel_launch(void* const* d_in, const int* in_sizes, int n_in,
                              void* d_out, int out_size, void* d_ws, size_t ws_size,
                              hipStream_t stream) {
  const float* x    = (const float*)d_in[0];   // [4, 2048, 1024]
  const float* W    = (const float*)d_in[1];   // [50257, 1024]
  const float* bias = (const float*)d_in[2];   // [50257]
  float* out = (float*)d_out;                  // [4, 50257]

  float* part = (float*)d_ws;                  // 32 * 4096 floats = 512 KB
  float* xsum = part + 32 * (BATCH * D_MODEL); // 4096 floats

  lmhead_reduce_partial<<<dim3(4, 32, 4), 256, 0, stream>>>(x, part);
  lmhead_combine<<<dim3(16), 256, 0, stream>>>(part, xsum);

  const int blocks = (NTILES + 7) / 8;         // 8 waves (tiles) per block
  lmhead_gemv_wmma<<<dim3(blocks), 256, 0, stream>>>(W, bias, xsum, out);
}